// Relu_5866925326756
// MI455X (gfx1250) — compile-verified
//
#include <hip/hip_runtime.h>

// Elementwise: |x| < EPS  ->  (x < 0 ? -EPS : +EPS), else x.
// Pure HBM-streaming workload (537 MB traffic, ~23 us floor at 23.3 TB/s).
// Hot kernel: branch-free, 32-bit indexed, 4x global_load_b128/store_b128
// with non-temporal hints. Tail handled by a separate (normally unlaunched)
// scalar kernel so the hot path carries zero control flow.

#define EPS_VAL 0.3f

typedef __attribute__((ext_vector_type(4))) float v4f;

__device__ __forceinline__ float eps_push(float v) {
    // matches jnp: where(|x| < eps, where(x < 0, -eps, eps), x)
    // (-0.0f < 0.0f) is false -> +eps, same as the reference.
    float s = (v < 0.0f) ? -EPS_VAL : EPS_VAL;
    return (__builtin_fabsf(v) < EPS_VAL) ? s : v;
}

__device__ __forceinline__ v4f eps_push4(v4f v) {
    v4f r;
    r.x = eps_push(v.x);
    r.y = eps_push(v.y);
    r.z = eps_push(v.z);
    r.w = eps_push(v.w);
    return r;
}

// Main kernel: each block consumes 1024 contiguous v4f chunks (16 KiB).
// Each thread handles 4 chunks strided by 256 -> every load/store is fully
// coalesced across the wave (wave32 x 16B = 512B per VMEM op).
// No bounds check: launcher only covers the exactly-divisible region.
__global__ __launch_bounds__(256) void eps_clamp_main_kernel(
    const v4f* __restrict__ x, v4f* __restrict__ y) {
    unsigned base = blockIdx.x * 1024u + threadIdx.x;

    v4f v0 = __builtin_nontemporal_load(x + base);
    v4f v1 = __builtin_nontemporal_load(x + base + 256u);
    v4f v2 = __builtin_nontemporal_load(x + base + 512u);
    v4f v3 = __builtin_nontemporal_load(x + base + 768u);

    v0 = eps_push4(v0);
    v1 = eps_push4(v1);
    v2 = eps_push4(v2);
    v3 = eps_push4(v3);

    __builtin_nontemporal_store(v0, y + base);
    __builtin_nontemporal_store(v1, y + base + 256u);
    __builtin_nontemporal_store(v2, y + base + 512u);
    __builtin_nontemporal_store(v3, y + base + 768u);
}

// Tail kernel: one block, grid-stride scalar over [start, n).
// Never launched for 8192x8192 (n % 4096 == 0), kept for generality.
__global__ __launch_bounds__(256) void eps_clamp_tail_kernel(
    const float* __restrict__ x, float* __restrict__ y,
    long long start, long long n) {
    for (long long j = start + threadIdx.x; j < n; j += 256) {
        y[j] = eps_push(x[j]);
    }
}

extern "C" void kernel_launch(void* const* d_in, const int* in_sizes, int n_in,
                              void* d_out, int out_size, void* d_ws, size_t ws_size,
                              hipStream_t stream) {
    (void)n_in; (void)d_ws; (void)ws_size; (void)out_size;
    const float* x = (const float*)d_in[0];
    float* y = (float*)d_out;
    long long n = (long long)in_sizes[0];   // 8192*8192 = 67,108,864

    // Each main block covers 1024 v4f = 4096 floats.
    long long mainBlocks = n / 4096;                 // 16,384 blocks
    long long covered    = mainBlocks * 4096;        // == n for this shape

    if (mainBlocks > 0) {
        eps_clamp_main_kernel<<<(dim3)(unsigned)mainBlocks, 256, 0, stream>>>(
            (const v4f*)x, (v4f*)y);
    }
    if (covered < n) {
        eps_clamp_tail_kernel<<<1, 256, 0, stream>>>(x, y, covered, n);
    }
}